// KMeansPlusPlusInitInitializer_78125455114499
// MI455X (gfx1250) — compile-verified
//
#include <hip/hip_runtime.h>

// k-means++ init for MI455X (gfx1250, wave32).
// Memory-bound, L2-resident (64MB buffer < 192MB L2). Distances via
// d(x,c) = |x|^2 - 2 x.c + |c|^2 with x.c computed by V_WMMA_F32_16X16X4_F32
// (fp32 WMMA, K=4 chunks x 16 = 64 dims), B = centroid broadcast over columns.

typedef __attribute__((ext_vector_type(2))) float v2f;
typedef __attribute__((ext_vector_type(8))) float v8f;

#define N_POINTS 262144
#define DIM 64
#define N_CLUSTERS 256
#define UPD_BLOCK 256                               // 8 waves of 32
#define POINTS_PER_BLOCK 128                        // 16 points per wave
#define N_UPD_BLOCKS (N_POINTS / POINTS_PER_BLOCK)  // 2048

__device__ __forceinline__ unsigned hash_u32(unsigned x) {
  x ^= x >> 16; x *= 0x7feb352dU;
  x ^= x >> 15; x *= 0x846ca68bU;
  x ^= x >> 16;
  return x;
}
__device__ __forceinline__ float hash_unit(unsigned x) {
  return (float)(hash_u32(x) >> 8) * (1.0f / 16777216.0f);  // [0,1)
}

// ---------------------------------------------------------------------------
// Pass 0: per-point squared norms, init min_d.
// ---------------------------------------------------------------------------
__global__ __launch_bounds__(256) void precompute_kernel(
    const float* __restrict__ buf, float* __restrict__ xnorm,
    float* __restrict__ min_d) {
  int p = blockIdx.x * blockDim.x + threadIdx.x;
  if (p >= N_POINTS) return;
  const float* row = buf + (size_t)p * DIM;
  float s = 0.f;
#pragma unroll
  for (int i = 0; i < DIM; i += 4) {
    float4 v = *reinterpret_cast<const float4*>(row + i);
    s += v.x * v.x + v.y * v.y + v.z * v.z + v.w * v.w;
  }
  xnorm[p] = s;
  min_d[p] = 3.402823466e38f;
}

// ---------------------------------------------------------------------------
// Per-step streaming pass: min_d = min(min_d, d(x, c)) using WMMA dot
// products; emit per-block sums of updated min_d for weighted sampling.
// One wave handles 16 points; 16 chunks of K=4 accumulate 64-dim dots.
// ---------------------------------------------------------------------------
__global__ __launch_bounds__(UPD_BLOCK) void update_kernel(
    const float* __restrict__ buf, const float* __restrict__ xnorm,
    const float* __restrict__ curc, const float* __restrict__ cnormp,
    float* __restrict__ min_d, float* __restrict__ blockSums) {
  __shared__ float s_c[DIM];
  __shared__ float s_dots[8][16];
  __shared__ float s_sum;

  const int tid = threadIdx.x;
  if (tid < DIM) s_c[tid] = curc[tid];
  if (tid == 0) s_sum = 0.f;
  __syncthreads();

  const int wave = tid >> 5;
  const int lane = tid & 31;
  const int hi = lane >> 4;   // 0: K={0,1}, 1: K={2,3} within a chunk
  const int lo = lane & 15;   // point row within the wave's 16-point tile
  const int pbase = blockIdx.x * POINTS_PER_BLOCK + wave * 16;
  const float* arow = buf + (size_t)(pbase + lo) * DIM + 2 * hi;

  v8f acc = {};
#pragma unroll
  for (int k = 0; k < 16; ++k) {
    const int col = 4 * k;
    v2f a, b;
    a.x = arow[col];
    a.y = arow[col + 1];
    b.x = s_c[col + 2 * hi];      // centroid chunk, broadcast over N columns
    b.y = s_c[col + 2 * hi + 1];
    // D[m][n] = sum_k A[m][k] * c[k]  (all 16 columns identical)
    acc = __builtin_amdgcn_wmma_f32_16x16x4_f32(
        false, a, false, b, (short)0, acc, false, false);
  }

  // D layout: lanes 0-15 hold M=0..7 in acc[0..7]; lanes 16-31 hold M=8..15.
  if (lane == 0 || lane == 16) {
#pragma unroll
    for (int v = 0; v < 8; ++v) s_dots[wave][hi * 8 + v] = acc[v];
  }
  __syncthreads();

  if (lane < 16) {
    const int p = pbase + lane;
    float d = xnorm[p] - 2.0f * s_dots[wave][lane] + cnormp[0];
    d = fmaxf(d, 0.0f);
    float nd = fminf(min_d[p], d);
    min_d[p] = nd;
    atomicAdd(&s_sum, nd);  // LDS float atomic
  }
  __syncthreads();
  if (tid == 0) blockSums[blockIdx.x] = s_sum;
}

// ---------------------------------------------------------------------------
// Selection: weighted sample proportional to min_d (categorical), fully
// parallel two-level prefix scan (2048 block sums -> 128 in-block values),
// then copy the chosen point as the next centroid and compute |c|^2.
// ---------------------------------------------------------------------------
__global__ __launch_bounds__(256) void select_kernel(
    const float* __restrict__ buf, const float* __restrict__ min_d,
    const float* __restrict__ blockSums, float* __restrict__ centroids,
    float* __restrict__ curc, float* __restrict__ cnormp, int iter) {
  __shared__ float sc[256];
  __shared__ float s_local[256];
  __shared__ int s_bsel;
  __shared__ float s_cumbase;
  __shared__ int s_idx;
  const int t = threadIdx.x;

  if (iter == 0) {
    if (t == 0) s_idx = (int)(hash_u32(0x9E3779B9u) % N_POINTS);
    __syncthreads();
  } else {
    // ---- level 1: locate the 128-point block containing the target ----
    float ls = 0.f;
#pragma unroll
    for (int j = 0; j < 8; ++j) ls += blockSums[t * 8 + j];
    s_local[t] = ls;
    sc[t] = ls;
    if (t == 0) { s_bsel = 0; s_cumbase = 0.f; s_idx = N_POINTS - 1; }
    __syncthreads();
    for (int off = 1; off < 256; off <<= 1) {  // inclusive Hillis-Steele scan
      float add = (t >= off) ? sc[t - off] : 0.f;
      __syncthreads();
      sc[t] += add;
      __syncthreads();
    }
    const float total = sc[255];
    const float u = hash_unit((unsigned)iter * 2654435761u + 0x85ebca6bu);
    const float target = u * total;

    const float incl = sc[t];
    const float excl = incl - ls;
    bool win = (target >= excl) && (target < incl);
    if (t == 255 && target >= total) win = true;  // rounding fallback
    if (win) {
      float cum = excl;
      int bsel = t * 8 + 7;
      for (int j = 0; j < 8; ++j) {
        const float s = blockSums[t * 8 + j];
        if (target < cum + s) { bsel = t * 8 + j; break; }
        cum += s;
      }
      s_bsel = bsel;
      s_cumbase = cum;
    }
    __syncthreads();

    // ---- level 2: locate the exact point inside the chosen block ----
    const int bsel = s_bsel;
    const float cumbase = s_cumbase;
    float v = 0.f;
    if (t < POINTS_PER_BLOCK) v = min_d[bsel * POINTS_PER_BLOCK + t];
    sc[t] = (t < POINTS_PER_BLOCK) ? v : 0.f;
    __syncthreads();
    for (int off = 1; off < 256; off <<= 1) {
      float add = (t >= off) ? sc[t - off] : 0.f;
      __syncthreads();
      sc[t] += add;
      __syncthreads();
    }
    if (t == 0) s_idx = bsel * POINTS_PER_BLOCK + POINTS_PER_BLOCK - 1;
    __syncthreads();
    if (t < POINTS_PER_BLOCK) {
      const float incl2 = cumbase + sc[t];
      const float excl2 = incl2 - v;
      if (v > 0.f && target >= excl2 && target < incl2)
        s_idx = bsel * POINTS_PER_BLOCK + t;
    }
    __syncthreads();
  }

  // ---- emit centroid, stash current centroid + its squared norm ----
  const int idx = s_idx;
  if (t < DIM) {
    const float v = buf[(size_t)idx * DIM + t];
    centroids[(size_t)iter * DIM + t] = v;
    curc[t] = v;
    sc[t] = v * v;
  }
  __syncthreads();
  if (t == 0) {
    float s = 0.f;
    for (int i = 0; i < DIM; ++i) s += sc[i];
    *cnormp = s;
  }
}

// ---------------------------------------------------------------------------
extern "C" void kernel_launch(void* const* d_in, const int* in_sizes, int n_in,
                              void* d_out, int out_size, void* d_ws,
                              size_t ws_size, hipStream_t stream) {
  const float* buf = (const float*)d_in[0];
  float* out = (float*)d_out;  // 256 x 64 f32

  // ws layout (~2.1 MB): min_d | xnorm | blockSums | curc | cnorm
  float* min_d = (float*)d_ws;
  float* xnorm = min_d + N_POINTS;
  float* blockSums = xnorm + N_POINTS;
  float* curc = blockSums + N_UPD_BLOCKS;
  float* cnorm = curc + DIM;

  precompute_kernel<<<N_POINTS / 256, 256, 0, stream>>>(buf, xnorm, min_d);
  select_kernel<<<1, 256, 0, stream>>>(buf, min_d, blockSums, out, curc,
                                       cnorm, 0);
  for (int i = 1; i < N_CLUSTERS; ++i) {
    update_kernel<<<N_UPD_BLOCKS, UPD_BLOCK, 0, stream>>>(
        buf, xnorm, curc, cnorm, min_d, blockSums);
    select_kernel<<<1, 256, 0, stream>>>(buf, min_d, blockSums, out, curc,
                                         cnorm, i);
  }
}